// SumMatrix_1649267441713
// MI455X (gfx1250) — compile-verified
//
#include <hip/hip_runtime.h>
#include <math.h>

// Problem constants (reference: B=16, M=2048, H=128)
#define BB 16
#define MM 2048
#define HH 128

typedef __attribute__((ext_vector_type(2))) float v2f;
typedef __attribute__((ext_vector_type(8))) float v8f;

// ---------------------------------------------------------------------------
// Kernel 1: s[b,m] = dot(h[b,m,:], w_e)  via V_WMMA_F32_16X16X4_F32.
// Grid: (B*M/128) blocks x 256 threads (8 waves); each wave does 16 rows,
// accumulating 32 WMMA steps of K=4 over H=128.
// A layout (16x4 f32): lane L<16 -> row L, holds K=k,k+1 ; lane L+16 -> row L,
// holds K=k+2,k+3. B broadcasts w[k..k+3] across all N columns, so every
// column of D equals s for the 16 rows.
// ---------------------------------------------------------------------------
__global__ __launch_bounds__(256) void k_rowdot_wmma(
    const float* __restrict__ h, const float* __restrict__ w,
    float* __restrict__ s) {
  __shared__ float wlds[HH];
  if (threadIdx.x < HH) wlds[threadIdx.x] = w[threadIdx.x];
  __syncthreads();

  const int wave = threadIdx.x >> 5;
  const int lane = threadIdx.x & 31;
  const int m    = lane & 15;
  const int hi   = lane >> 4;  // 0: K pair (k,k+1), 1: K pair (k+2,k+3)

  const long long row0 = (long long)blockIdx.x * 128 + wave * 16;
  const float* hrow = h + (row0 + m) * HH;

  v8f c = {};
  for (int k = 0; k < HH; k += 4) {
    v2f a, b;
    a.x = hrow[k + 2 * hi];
    a.y = hrow[k + 2 * hi + 1];
    b.x = wlds[k + 2 * hi];
    b.y = wlds[k + 2 * hi + 1];
    c = __builtin_amdgcn_wmma_f32_16x16x4_f32(
        /*neg_a=*/false, a, /*neg_b=*/false, b,
        /*c_mod=*/(short)0, c, /*reuse_a=*/false, /*reuse_b=*/false);
  }
  // D layout: VGPR r: lanes 0-15 -> M=r, lanes 16-31 -> M=8+r (all N equal).
  if (m == 0) {
    for (int r = 0; r < 8; ++r) s[row0 + hi * 8 + r] = c[r];
  }
}

// ---------------------------------------------------------------------------
// Kernel 2: p[b,:] = softmax(s[b,:]).  Grid: B blocks x 256 threads.
// ---------------------------------------------------------------------------
__global__ __launch_bounds__(256) void k_softmax(
    const float* __restrict__ s, float* __restrict__ p) {
  __shared__ float red[256];
  const int b = blockIdx.x;
  const float* sb = s + (long long)b * MM;
  float* pb = p + (long long)b * MM;

  float loc[8];
  float mx = -INFINITY;
  for (int t = 0; t < 8; ++t) {
    loc[t] = sb[threadIdx.x + t * 256];
    mx = fmaxf(mx, loc[t]);
  }
  red[threadIdx.x] = mx;
  __syncthreads();
  for (int off = 128; off > 0; off >>= 1) {
    if (threadIdx.x < off)
      red[threadIdx.x] = fmaxf(red[threadIdx.x], red[threadIdx.x + off]);
    __syncthreads();
  }
  mx = red[0];
  __syncthreads();

  float sum = 0.0f;
  for (int t = 0; t < 8; ++t) {
    loc[t] = __expf(loc[t] - mx);
    sum += loc[t];
  }
  red[threadIdx.x] = sum;
  __syncthreads();
  for (int off = 128; off > 0; off >>= 1) {
    if (threadIdx.x < off) red[threadIdx.x] += red[threadIdx.x + off];
    __syncthreads();
  }
  const float rs = 1.0f / red[0];
  for (int t = 0; t < 8; ++t) pb[threadIdx.x + t * 256] = loc[t] * rs;
}

// ---------------------------------------------------------------------------
// Kernel 3: Z[b,i] = sum_j p[b,j]*mask[b,i,j] + 1e-10.
// Pure streaming GEMV over mask (256 MB) — bandwidth bound, so fully
// coalesced float4 loads, p staged in LDS.  Grid: (M/64, B) x 256 threads;
// each wave scans 8 full rows (512 B per wave per load).
// ---------------------------------------------------------------------------
__global__ __launch_bounds__(256) void k_zrow(
    const float* __restrict__ mask, const float* __restrict__ p,
    float* __restrict__ Z) {
  __shared__ float pl[MM];
  const int b = blockIdx.y;
  const int tid = threadIdx.x;

  const float4* pg = (const float4*)(p + (long long)b * MM);
  for (int t = tid; t < MM / 4; t += 256) ((float4*)pl)[t] = pg[t];
  __syncthreads();

  const int wave = tid >> 5, lane = tid & 31;
  const long long base =
      ((long long)b * MM + (long long)blockIdx.x * 64) * MM;

  for (int rr = 0; rr < 8; ++rr) {
    const int row = wave * 8 + rr;
    const float4* mrow = (const float4*)(mask + base + (long long)row * MM);
    const float4* pv4 = (const float4*)pl;
    float acc = 0.0f;
    for (int it = 0; it < MM / 128; ++it) {
      float4 mv = mrow[lane + it * 32];
      float4 pv = pv4[lane + it * 32];
      acc += mv.x * pv.x + mv.y * pv.y + mv.z * pv.z + mv.w * pv.w;
    }
    for (int off = 16; off > 0; off >>= 1) acc += __shfl_xor(acc, off, 32);
    if (lane == 0)
      Z[(long long)b * MM + blockIdx.x * 64 + row] = acc + 1e-10f;
  }
}

// ---------------------------------------------------------------------------
// Kernel 4: out[b,i,j] = 0.5*(p_j*mask_ij/Z_i + p_i*mask_ji/Z_j).
// Result is symmetric, so handle tile pairs (ti<=tj): read mask tiles
// (ti,tj) and (tj,ti) once (both coalesced), write out tile (ti,tj)
// coalesced and out tile (tj,ti) via an LDS transpose (also coalesced).
// Total traffic: 256 MB mask read + 256 MB out write.
// Grid: (32, 32, B) x 256 threads, 64x64 tiles; upper-triangle blocks only.
// ---------------------------------------------------------------------------
__global__ __launch_bounds__(256) void k_out(
    const float* __restrict__ mask, const float* __restrict__ p,
    const float* __restrict__ Z, float* __restrict__ out) {
  const int ti = blockIdx.x, tj = blockIdx.y, b = blockIdx.z;
  if (tj < ti) return;

  __shared__ float s2[64][65];   // mask[b, j0+rr, i0+cc]
  __shared__ float so[64][65];   // out tile1 staged for transposed write
  __shared__ float pi[64], pj[64], rZi[64], rZj[64];

  const int tid = threadIdx.x;
  const long long mb = (long long)b * MM * MM;
  const int i0 = ti * 64, j0 = tj * 64;

  if (tid < 64) {
    pi[tid]  = p[b * MM + i0 + tid];
    rZi[tid] = 1.0f / Z[b * MM + i0 + tid];
  } else if (tid < 128) {
    const int t = tid - 64;
    pj[t]  = p[b * MM + j0 + t];
    rZj[t] = 1.0f / Z[b * MM + j0 + t];
  }

  const int c4 = tid & 15, r0 = tid >> 4;
  for (int rb = 0; rb < 4; ++rb) {
    const int rr = r0 + rb * 16;
    float4 mv = *(const float4*)(mask + mb + (long long)(j0 + rr) * MM +
                                 i0 + c4 * 4);
    s2[rr][c4 * 4 + 0] = mv.x;
    s2[rr][c4 * 4 + 1] = mv.y;
    s2[rr][c4 * 4 + 2] = mv.z;
    s2[rr][c4 * 4 + 3] = mv.w;
  }
  __syncthreads();

  for (int rb = 0; rb < 4; ++rb) {
    const int r = r0 + rb * 16;
    const int c = c4 * 4;
    float4 m1 = *(const float4*)(mask + mb + (long long)(i0 + r) * MM +
                                 j0 + c);
    float4 v;
    v.x = 0.5f * (pj[c + 0] * m1.x * rZi[r] + pi[r] * s2[c + 0][r] * rZj[c + 0]);
    v.y = 0.5f * (pj[c + 1] * m1.y * rZi[r] + pi[r] * s2[c + 1][r] * rZj[c + 1]);
    v.z = 0.5f * (pj[c + 2] * m1.z * rZi[r] + pi[r] * s2[c + 2][r] * rZj[c + 2]);
    v.w = 0.5f * (pj[c + 3] * m1.w * rZi[r] + pi[r] * s2[c + 3][r] * rZj[c + 3]);
    *(float4*)(out + mb + (long long)(i0 + r) * MM + j0 + c) = v;
    so[r][c + 0] = v.x;
    so[r][c + 1] = v.y;
    so[r][c + 2] = v.z;
    so[r][c + 3] = v.w;
  }

  if (ti == tj) return;  // diagonal tile: single symmetric write covers all
  __syncthreads();

  // out[b, j0+rr, i0+cc] = so[cc][rr] (out is symmetric)
  for (int rb = 0; rb < 4; ++rb) {
    const int rr = r0 + rb * 16;
    const int cc = c4 * 4;
    float4 v;
    v.x = so[cc + 0][rr];
    v.y = so[cc + 1][rr];
    v.z = so[cc + 2][rr];
    v.w = so[cc + 3][rr];
    *(float4*)(out + mb + (long long)(j0 + rr) * MM + i0 + cc) = v;
  }
}

// ---------------------------------------------------------------------------
extern "C" void kernel_launch(void* const* d_in, const int* in_sizes, int n_in,
                              void* d_out, int out_size, void* d_ws,
                              size_t ws_size, hipStream_t stream) {
  const float* h    = (const float*)d_in[0];
  const float* mask = (const float*)d_in[1];
  const float* w_e  = (const float*)d_in[2];
  // d_in[3] (b_e) cancels inside the row softmax -> unused.
  float* out = (float*)d_out;

  // Workspace layout: p [B*M], Z [B*M], s [B*M]  (384 KB total)
  float* p = (float*)d_ws;
  float* Z = p + (size_t)BB * MM;
  float* s = Z + (size_t)BB * MM;

  k_rowdot_wmma<<<dim3(BB * MM / 128), 256, 0, stream>>>(h, w_e, s);
  k_softmax<<<dim3(BB), 256, 0, stream>>>(s, p);
  k_zrow<<<dim3(MM / 64, BB), 256, 0, stream>>>(mask, p, Z);
  k_out<<<dim3(MM / 64, MM / 64, BB), 256, 0, stream>>>(mask, p, Z, out);
}